// GCN_7215545057921
// MI455X (gfx1250) — compile-verified
//
#include <hip/hip_runtime.h>

typedef __attribute__((ext_vector_type(2))) float v2f;
typedef __attribute__((ext_vector_type(8))) float v8f;

#define F_IN 512
#define F1 128
#define F2 64

// ---------------- degree / norm ----------------

__global__ void init_deg_kernel(float* __restrict__ deg, int n) {
  int i = blockIdx.x * blockDim.x + threadIdx.x;
  if (i < n) deg[i] = 1.0f;  // self-loop contributes 1 to every node's degree
}

__global__ void deg_accum_kernel(const int* __restrict__ dst, float* __restrict__ deg, int e) {
  int i = blockIdx.x * blockDim.x + threadIdx.x;
  if (i < e) atomicAdd(&deg[dst[i]], 1.0f);
}

__global__ void rsqrt_kernel(float* __restrict__ deg, int n) {
  int i = blockIdx.x * blockDim.x + threadIdx.x;
  if (i < n) deg[i] = rsqrtf(deg[i]);  // deg >= 1 always (self-loop), no zero guard needed
}

__global__ void zero_kernel(float* __restrict__ p, int n) {
  int i = blockIdx.x * blockDim.x + threadIdx.x;
  if (i < n) p[i] = 0.0f;
}

// ---------------- fp32 WMMA GEMM: C[M,N] = A[M,K] @ B[K,N], all row-major ----------------
// One block handles a 16-row stripe of A and ALL N columns.
// N/16 waves per block; wave w computes the 16x16 tile at columns [16w, 16w+16).
// Requires M % 16 == 0 (holds: 100000 = 6250*16).
//
// V_WMMA_F32_16X16X4_F32 operand layout (ISA 7.12.2, 32-bit A 16x4):
//   lane l in [0,16):  a.v0 = A[M=l][K=0], a.v1 = A[M=l][K=1]
//   lane l+16:         a.v0 = A[M=l][K=2], a.v1 = A[M=l][K=3]
// B (4x16) is the mirror: row striped across lanes within a VGPR:
//   lane l in [0,16):  b.v0 = B[K=0][N=l], b.v1 = B[K=1][N=l]
//   lane l+16:         b.v0 = B[K=2][N=l], b.v1 = B[K=3][N=l]
// C/D (16x16 f32, 8 VGPRs): vgpr r, lane l(<16) -> C[M=r][N=l]; lane l+16 -> C[M=r+8][N=l]
template <int K, int N>
__global__ __launch_bounds__(32 * (N / 16)) void gemm_wmma_kernel(
    const float* __restrict__ A, const float* __restrict__ B, float* __restrict__ C) {
  constexpr int NTHREADS = 32 * (N / 16);
  __shared__ float lds[16 * K];

  const int row0 = blockIdx.x * 16;
  const int tid = threadIdx.x;

  // The 16 x K tile of row-major A is one contiguous span (full-K rows).
  const float4* __restrict__ src4 = reinterpret_cast<const float4*>(A + (size_t)row0 * K);
  float4* lds4 = reinterpret_cast<float4*>(lds);
  constexpr int COPIES = (16 * K / 4) / NTHREADS;
#pragma unroll
  for (int i = 0; i < COPIES; ++i) lds4[tid + i * NTHREADS] = src4[tid + i * NTHREADS];
  __syncthreads();

  const int wave = tid >> 5;
  const int lane = tid & 31;
  const int half = lane >> 4;  // selects K-pair {0,1} vs {2,3} within the 16x4 fragment
  const int l = lane & 15;
  const int col0 = wave * 16;

  v8f c = {};
#pragma unroll 8
  for (int kk = 0; kk < K; kk += 4) {
    const int ka = kk + 2 * half;
    v2f a;
    a.x = lds[l * K + ka];      // consecutive in LDS -> single ds_load_b64
    a.y = lds[l * K + ka + 1];
    v2f b;
    b.x = B[(size_t)ka * N + col0 + l];        // W is small, L2/L0 resident
    b.y = B[(size_t)(ka + 1) * N + col0 + l];
    // 8 args: (neg_a, A, neg_b, B, c_mod, C, reuse_a, reuse_b)
    c = __builtin_amdgcn_wmma_f32_16x16x4_f32(false, a, false, b, (short)0, c, false, false);
  }

#pragma unroll
  for (int r = 0; r < 8; ++r) {
    C[(size_t)(row0 + r + 8 * half) * N + col0 + l] = c[r];
  }
}

// ---------------- edge scatter: accum[dst] += feat[src] * dinv[src]*dinv[dst] ----------------
// One wave per edge; each lane handles F/32 consecutive floats.
// feat/accum (<= 51 MB each) are L2-resident, so the gathers + f32 atomics stay on-chip.
template <int F>
__global__ void scatter_kernel(const int* __restrict__ src, const int* __restrict__ dst,
                               const float* __restrict__ dinv, const float* __restrict__ feat,
                               float* __restrict__ accum, int e) {
  constexpr int PER = F / 32;
  const int gid = blockIdx.x * blockDim.x + threadIdx.x;
  const int edge = gid >> 5;
  const int lane = gid & 31;
  if (edge >= e) return;
  const int s = src[edge];
  const int d = dst[edge];
  const float norm = dinv[s] * dinv[d];
  const float* fs = feat + (size_t)s * F + lane * PER;
  float* out = accum + (size_t)d * F + lane * PER;
  if constexpr (PER == 4) {
    float4 v = *reinterpret_cast<const float4*>(fs);
    atomicAdd(out + 0, v.x * norm);
    atomicAdd(out + 1, v.y * norm);
    atomicAdd(out + 2, v.z * norm);
    atomicAdd(out + 3, v.w * norm);
  } else {
    float2 v = *reinterpret_cast<const float2*>(fs);
    atomicAdd(out + 0, v.x * norm);
    atomicAdd(out + 1, v.y * norm);
  }
}

// ---------------- finalize: accum += feat * dinv^2 (self-loop) + bias; optional ReLU ----------------
template <bool RELU, int F>
__global__ void finalize_kernel(float* __restrict__ accum, const float* __restrict__ feat,
                                const float* __restrict__ dinv, const float* __restrict__ bias,
                                int n) {
  const int i = blockIdx.x * blockDim.x + threadIdx.x;
  if (i >= n * F) return;
  const int node = i / F;  // F is a power of two -> shift
  const int f = i & (F - 1);
  const float di = dinv[node];
  float v = accum[i] + feat[i] * di * di + bias[f];
  accum[i] = RELU ? fmaxf(v, 0.0f) : v;
}

// ---------------- driver ----------------

extern "C" void kernel_launch(void* const* d_in, const int* in_sizes, int n_in,
                              void* d_out, int out_size, void* d_ws, size_t ws_size,
                              hipStream_t stream) {
  const float* x  = (const float*)d_in[0];
  const int*   ei = (const int*)d_in[1];
  const float* W1 = (const float*)d_in[2];
  const float* b1 = (const float*)d_in[3];
  const float* W2 = (const float*)d_in[4];
  const float* b2 = (const float*)d_in[5];
  float* out = (float*)d_out;

  const int n = in_sizes[0] / F_IN;  // 100000 (multiple of 16)
  const int e = in_sizes[1] / 2;     // 1600000
  const int* src = ei;
  const int* dst = ei + e;

  // workspace layout (floats): xw[n*128] | acc1/h[n*128] | hw[n*64] | dinv[n]
  float* ws   = (float*)d_ws;
  float* xw   = ws;
  float* acc1 = xw + (size_t)n * F1;
  float* hw   = acc1 + (size_t)n * F1;
  float* dinv = hw + (size_t)n * F2;

  const int T = 256;

  // degrees -> dinv (in place)
  init_deg_kernel<<<(n + T - 1) / T, T, 0, stream>>>(dinv, n);
  deg_accum_kernel<<<(e + T - 1) / T, T, 0, stream>>>(dst, dinv, e);
  rsqrt_kernel<<<(n + T - 1) / T, T, 0, stream>>>(dinv, n);

  // layer 1: xw = x @ W1  (M=n, K=512, N=128) -> 8 waves/block
  gemm_wmma_kernel<F_IN, F1><<<n / 16, 256, 0, stream>>>(x, W1, xw);
  zero_kernel<<<((n * F1) + T - 1) / T, T, 0, stream>>>(acc1, n * F1);
  scatter_kernel<F1><<<((e * 32) + T - 1) / T, T, 0, stream>>>(src, dst, dinv, xw, acc1, e);
  finalize_kernel<true, F1><<<((n * F1) + T - 1) / T, T, 0, stream>>>(acc1, xw, dinv, b1, n);

  // layer 2: hw = h @ W2  (M=n, K=128, N=64) -> 4 waves/block
  gemm_wmma_kernel<F1, F2><<<n / 16, 128, 0, stream>>>(acc1, W2, hw);
  zero_kernel<<<((n * F2) + T - 1) / T, T, 0, stream>>>(out, n * F2);
  scatter_kernel<F2><<<((e * 32) + T - 1) / T, T, 0, stream>>>(src, dst, dinv, hw, out, e);
  finalize_kernel<false, F2><<<((n * F2) + T - 1) / T, T, 0, stream>>>(out, hw, dinv, b2, n);
}